// ResGATLayer_13864154432005
// MI455X (gfx1250) — compile-verified
//
#include <hip/hip_runtime.h>
#include <hip/hip_bf16.h>

#define IN_DIM    256   // also HEADS*OUT_DIM
#define OUT_DIM   64
#define HEADS     4
#define NEG_SLOPE 0.2f

typedef __attribute__((ext_vector_type(2))) float v2f;
typedef __attribute__((ext_vector_type(8))) float v8f;
typedef __attribute__((ext_vector_type(4))) unsigned int v4u;
typedef __attribute__((ext_vector_type(8))) int v8i;
typedef __attribute__((ext_vector_type(4))) int v4i;

__device__ __forceinline__ float leaky(float v) {
    return v > 0.0f ? v : NEG_SLOPE * v;
}

__device__ __forceinline__ void atomicMaxF32(float* addr, float val) {
    // sign-magnitude trick: non-negative floats order as ints,
    // negative floats order reversed as unsigned. Init must be -inf bits.
    if (val >= 0.0f) {
        atomicMax((int*)addr, __float_as_int(val));
    } else {
        atomicMin((unsigned int*)addr, (unsigned int)__float_as_int(val));
    }
}

// ---------------------------------------------------------------------------
// Kernel 1: xh = x @ W   ([N,256] x [256,256] -> [N,256])
// block = 128 threads = 4 waves. Each wave: 16 rows x 64 cols (one head).
// The 16x256 A tile (16 KB, contiguous in x) is DMA'd into LDS once per block
// by the Tensor Data Mover; all 4 waves then read A fragments from LDS
// (ds_load_b64) while streaming B (L2-resident W) from global.
// A 16x4 frag: lane L (m=L&15, half=L>>4), VGPR v -> element (m, k + 2*half + v)
// B 4x16 frag: VGPR v -> row K = k + 2*half + v, col = m
// C/D 16x16:   VGPR r -> row (r + 8*half), col = m
// ---------------------------------------------------------------------------
__global__ void gat_gemm_wmma(const float* __restrict__ X,
                              const float* __restrict__ Wm,
                              float* __restrict__ XH, int N) {
    __shared__ float As[16 * IN_DIM];            // 16 KB A tile

    const int lane = threadIdx.x & 31;
    const int wid  = threadIdx.x >> 5;           // 0..3 : 64-col strip == head
    const int m    = lane & 15;
    const int half = lane >> 4;
    const int row0 = blockIdx.x * 16;
    const int colw = wid * OUT_DIM;

    // ---- TDM: DMA the A tile (rows [row0,row0+16) x all 256 cols) into LDS.
    if (wid == 0) {
        int rows = N - row0;
        if (rows > 16) rows = 16;
        const unsigned long long gaddr =
            (unsigned long long)(const void*)(X + (size_t)row0 * IN_DIM);
        const unsigned int laddr = (unsigned int)(unsigned long long)(void*)As;

        v4u g0;
        g0[0] = 1u;                                            // count=1, user D#
        g0[1] = laddr;                                         // lds_addr [63:32]
        g0[2] = (unsigned int)(gaddr & 0xFFFFFFFFull);         // global_addr lo
        g0[3] = (unsigned int)((gaddr >> 32) & 0x1FFFFFFull)   // global_addr hi
              | (2u << 30);                                    // type = 2 (image)

        v8i g1;
        g1[0] = (int)(2u << 16);                 // data_size=4B, mask=0, no flags
        g1[1] = (int)(256u << 16);               // tensor_dim0[15:0]=256 @bits63:48
        g1[2] = (int)(((unsigned)rows) << 16);   // dim0 hi=0 | tensor_dim1 lo=rows
        g1[3] = (int)(256u << 16);               // dim1 hi=0 | tile_dim0=256
        g1[4] = 16;                              // tile_dim1=16 | tile_dim2=0
        g1[5] = 256;                             // tensor_dim0_stride lo = 256
        g1[6] = 0;                               // stride hi | dim1_stride lo
        g1[7] = 0;                               // dim1_stride hi
        const v4i gz  = {0, 0, 0, 0};            // 2-D tensor: groups 2/3 unused
        const v8i gz8 = {0, 0, 0, 0, 0, 0, 0, 0};

        __builtin_amdgcn_tensor_load_to_lds(g0, g1, gz, gz, gz8, 0);
        __builtin_amdgcn_s_wait_tensorcnt(0);
    }
    __syncthreads();

    v8f acc0 = {}, acc1 = {}, acc2 = {}, acc3 = {};

    for (int k = 0; k < IN_DIM; k += 4) {
        const int ka = k + 2 * half;
        v2f a;
        a.x = As[m * IN_DIM + ka + 0];           // ds_load_b64
        a.y = As[m * IN_DIM + ka + 1];

        const float* w0 = Wm + (size_t)(ka + 0) * IN_DIM + colw + m;
        const float* w1 = Wm + (size_t)(ka + 1) * IN_DIM + colw + m;
        v2f b0; b0.x = w0[0];  b0.y = w1[0];
        v2f b1; b1.x = w0[16]; b1.y = w1[16];
        v2f b2; b2.x = w0[32]; b2.y = w1[32];
        v2f b3; b3.x = w0[48]; b3.y = w1[48];

        if (k + 16 < IN_DIM)
            __builtin_prefetch(w0 + 16 * IN_DIM, 0, 1);

        acc0 = __builtin_amdgcn_wmma_f32_16x16x4_f32(false, a, false, b0, (short)0, acc0, false, false);
        acc1 = __builtin_amdgcn_wmma_f32_16x16x4_f32(false, a, false, b1, (short)0, acc1, false, false);
        acc2 = __builtin_amdgcn_wmma_f32_16x16x4_f32(false, a, false, b2, (short)0, acc2, false, false);
        acc3 = __builtin_amdgcn_wmma_f32_16x16x4_f32(false, a, false, b3, (short)0, acc3, false, false);
    }

#pragma unroll
    for (int r = 0; r < 8; ++r) {
        const int row = row0 + r + 8 * half;
        if (row < N) {
            float* o = XH + (size_t)row * IN_DIM + colw + m;
            o[0]  = acc0[r];
            o[16] = acc1[r];
            o[32] = acc2[r];
            o[48] = acc3[r];
        }
    }
}

// ---------------------------------------------------------------------------
// Kernel 2: per-node attention scores  a_src[n,h] = <xh[n,h,:], att_src[h,:]>
// One block (256 threads) per node; LDS tree-reduce per 64-channel head group.
// ---------------------------------------------------------------------------
__global__ void gat_att_scores(const float* __restrict__ XH,
                               const float* __restrict__ att_src,
                               const float* __restrict__ att_dst,
                               float* __restrict__ a_src,
                               float* __restrict__ a_dst) {
    __shared__ float ss[256];
    __shared__ float sd[256];
    const int n = blockIdx.x;
    const int c = threadIdx.x;
    const float v = XH[(size_t)n * IN_DIM + c];
    ss[c] = v * att_src[c];
    sd[c] = v * att_dst[c];
    __syncthreads();
#pragma unroll
    for (int off = 32; off >= 1; off >>= 1) {
        if ((c & 63) < off) { ss[c] += ss[c + off]; sd[c] += sd[c + off]; }
        __syncthreads();
    }
    if ((c & 63) == 0) {
        a_src[n * HEADS + (c >> 6)] = ss[c];
        a_dst[n * HEADS + (c >> 6)] = sd[c];
    }
}

// ---------------------------------------------------------------------------
// Init kernels (d_out / d_ws arrive poisoned; must set every call)
// ---------------------------------------------------------------------------
__global__ void gat_init_seg(float* __restrict__ smax, float* __restrict__ ssum, int n) {
    const int i = blockIdx.x * blockDim.x + threadIdx.x;
    if (i < n) { smax[i] = -__builtin_huge_valf(); ssum[i] = 0.0f; }
}

__global__ void gat_init_out(float* __restrict__ out, const float* __restrict__ bias) {
    const int n = blockIdx.x;
    const int c = threadIdx.x;
    out[(size_t)n * IN_DIM + c] = bias[c];
}

// ---------------------------------------------------------------------------
// Kernel 4: segment max of leaky-relu logits over destination nodes.
// One thread per (edge,head); edges [0,E) are real, [E,E+N) are self-loops.
// ---------------------------------------------------------------------------
__global__ void gat_edge_max(const int* __restrict__ ei,
                             const float* __restrict__ a_src,
                             const float* __restrict__ a_dst,
                             float* __restrict__ smax, int E, int N) {
    const long long idx = (long long)blockIdx.x * blockDim.x + threadIdx.x;
    if (idx >= (long long)(E + N) * HEADS) return;
    const int e = (int)(idx >> 2);
    const int h = (int)(idx & 3);
    int j, i;
    if (e < E) { j = ei[e]; i = ei[E + e]; } else { j = i = e - E; }
    const float l = leaky(a_src[j * HEADS + h] + a_dst[i * HEADS + h]);
    atomicMaxF32(&smax[i * HEADS + h], l);
}

// Kernel 5: segment sum of exp(logit - max)
__global__ void gat_edge_sum(const int* __restrict__ ei,
                             const float* __restrict__ a_src,
                             const float* __restrict__ a_dst,
                             const float* __restrict__ smax,
                             float* __restrict__ ssum, int E, int N) {
    const long long idx = (long long)blockIdx.x * blockDim.x + threadIdx.x;
    if (idx >= (long long)(E + N) * HEADS) return;
    const int e = (int)(idx >> 2);
    const int h = (int)(idx & 3);
    int j, i;
    if (e < E) { j = ei[e]; i = ei[E + e]; } else { j = i = e - E; }
    const float l = leaky(a_src[j * HEADS + h] + a_dst[i * HEADS + h]);
    const float v = __expf(l - smax[i * HEADS + h]);
    __hip_atomic_fetch_add(&ssum[i * HEADS + h], v,
                           __ATOMIC_RELAXED, __HIP_MEMORY_SCOPE_AGENT);
}

// ---------------------------------------------------------------------------
// Kernel 6: scatter-aggregate  out[i] += alpha * xh[j]
// One wave32 per edge: lanes 0..3 compute the 4 head alphas once, broadcast
// via __shfl; 8 iterations x 32 lanes cover 256 channels.
// ---------------------------------------------------------------------------
__global__ void gat_edge_scatter(const int* __restrict__ ei,
                                 const float* __restrict__ XH,
                                 const float* __restrict__ a_src,
                                 const float* __restrict__ a_dst,
                                 const float* __restrict__ smax,
                                 const float* __restrict__ ssum,
                                 float* __restrict__ out, int E, int N) {
    const int lane  = threadIdx.x & 31;
    const long long edge = ((long long)blockIdx.x * blockDim.x + threadIdx.x) >> 5;
    if (edge >= (long long)(E + N)) return;
    const int e = (int)edge;
    int j, i;
    if (e < E) { j = ei[e]; i = ei[E + e]; } else { j = i = e - E; }

    // per-head alpha in lanes 0..3 (all lanes compute a valid replica)
    const int h = lane & 3;
    const float l = leaky(a_src[j * HEADS + h] + a_dst[i * HEADS + h]);
    const float alpha = __expf(l - smax[i * HEADS + h]) / ssum[i * HEADS + h];

    const float* src = XH + (size_t)j * IN_DIM;
    float*       dst = out + (size_t)i * IN_DIM;
#pragma unroll
    for (int t = 0; t < 8; ++t) {
        const int c = t * 32 + lane;          // channel; head = t>>1
        const float a = __shfl(alpha, t >> 1, 32);
        __hip_atomic_fetch_add(&dst[c], a * src[c],
                               __ATOMIC_RELAXED, __HIP_MEMORY_SCOPE_AGENT);
    }
}

// ---------------------------------------------------------------------------
extern "C" void kernel_launch(void* const* d_in, const int* in_sizes, int n_in,
                              void* d_out, int out_size, void* d_ws, size_t ws_size,
                              hipStream_t stream) {
    (void)n_in; (void)out_size; (void)ws_size;
    const float* x    = (const float*)d_in[0];
    const int*   ei   = (const int*)d_in[1];
    const float* W    = (const float*)d_in[2];
    const float* asrc = (const float*)d_in[3];   // [H,C] flat == [256]
    const float* adst = (const float*)d_in[4];
    const float* bias = (const float*)d_in[5];
    float*       out  = (float*)d_out;

    const int N = in_sizes[0] / IN_DIM;          // 50000
    const int E = in_sizes[1] / 2;               // 800000

    float* ws     = (float*)d_ws;
    float* xh     = ws;                                   // N*256
    float* a_src  = xh    + (size_t)N * IN_DIM;           // N*4
    float* a_dst  = a_src + (size_t)N * HEADS;            // N*4
    float* smax   = a_dst + (size_t)N * HEADS;            // N*4
    float* ssum   = smax  + (size_t)N * HEADS;            // N*4

    // 1. xh = x @ W  (TDM-staged A tile + WMMA)
    gat_gemm_wmma<<<(N + 15) / 16, 128, 0, stream>>>(x, W, xh, N);
    // 2. attention scores
    gat_att_scores<<<N, 256, 0, stream>>>(xh, asrc, adst, a_src, a_dst);
    // 3. init segment stats + output(bias)
    gat_init_seg<<<(N * HEADS + 255) / 256, 256, 0, stream>>>(smax, ssum, N * HEADS);
    gat_init_out<<<N, 256, 0, stream>>>(out, bias);
    // 4/5. segment softmax over incoming edges
    const long long nt = (long long)(E + N) * HEADS;
    const int gmax = (int)((nt + 255) / 256);
    gat_edge_max<<<gmax, 256, 0, stream>>>(ei, a_src, a_dst, smax, E, N);
    gat_edge_sum<<<gmax, 256, 0, stream>>>(ei, a_src, a_dst, smax, ssum, E, N);
    // 6. weighted scatter aggregate (one wave per edge)
    const int gsc = (E + N + 7) / 8;             // 8 waves per 256-thread block
    gat_edge_scatter<<<gsc, 256, 0, stream>>>(ei, xh, a_src, a_dst, smax, ssum, out, E, N);
}